// DistanceNorm_86535001079919
// MI455X (gfx1250) — compile-verified
//
#include <hip/hip_runtime.h>

// Shapes fixed by the reference: distance (16, 2048, 1024) fp32.
#define B_  16
#define L_  2048
#define M_  1024
#define STRIPS      (M_ / 16)          // 64 column strips of 16
#define CHUNK_ROWS  128                // L-rows per wave
#define CHUNKS      (L_ / CHUNK_ROWS)  // 16
#define PARTIALS    (STRIPS * CHUNKS)  // 1024 partial slots per batch

typedef __attribute__((ext_vector_type(2))) float v2f;
typedef __attribute__((ext_vector_type(8))) float v8f;

// -------- Stage 1: per-wave column-sum reduction via V_WMMA_F32_16X16X4_F32 --
// Wave w owns (batch b, strip, chunk). A = ones(16x4), B = 4x16 tile of d.
// D accumulates column sums replicated in every row; acc[0] lane l holds
// colsum[colBase + l%16]. Butterfly-reduce lanes (both 16-lane halves hold the
// same values -> multiply by 0.5 after full reduce, which is exact).
__global__ __launch_bounds__(256)
void dn_colsum_wmma(const float* __restrict__ d, float* __restrict__ partials) {
    const int waveId = blockIdx.x * (blockDim.x / 32) + (threadIdx.x >> 5);
    const int lane   = threadIdx.x & 31;

    const int b     = waveId / PARTIALS;
    const int slot  = waveId % PARTIALS;
    const int strip = slot / CHUNKS;
    const int chunk = slot % CHUNKS;

    const int colBase = strip * 16;
    const int col     = colBase + (lane & 15);
    const int row0    = chunk * CHUNK_ROWS;
    // K-row pair held by this lane half (exact mapping irrelevant since A==1).
    const int krow    = (lane < 16) ? 0 : 2;

    const float* base = d + ((size_t)b * L_ + (size_t)(row0 + krow)) * M_ + col;

    v2f a; a[0] = 1.0f; a[1] = 1.0f;    // all-ones A matrix (16x4)
    v8f acc = {};

    #pragma unroll 4
    for (int kk = 0; kk < CHUNK_ROWS / 4; ++kk) {
        const float* p = base + (size_t)(4 * kk) * M_;
        v2f bt;
        bt[0] = p[0];      // K-row krow
        bt[1] = p[M_];     // K-row krow+1
        acc = __builtin_amdgcn_wmma_f32_16x16x4_f32(
            /*neg_a=*/false, a, /*neg_b=*/false, bt,
            /*c_mod=*/(short)0, acc, /*reuse_a=*/false, /*reuse_b=*/false);
    }

    float cs = acc[0];                       // colsum[col], duplicated per half
    float s0 = cs;
    float s1 = cs * (float)col;
    #pragma unroll
    for (int off = 16; off >= 1; off >>= 1) {
        s0 += __shfl_xor(s0, off, 32);
        s1 += __shfl_xor(s1, off, 32);
    }
    if (lane == 0) {
        float* slotp = partials + ((size_t)b * PARTIALS + slot) * 2;
        slotp[0] = s0 * 0.5f;   // both halves were identical -> exact halving
        slotp[1] = s1 * 0.5f;
    }
}

// -------- Stage 2: deterministic fixed-order reduction of partials -> scale --
__global__ __launch_bounds__(256)
void dn_finalize(const float* __restrict__ partials, float* __restrict__ scale) {
    __shared__ float sm0[256];
    __shared__ float sm1[256];
    const int b = blockIdx.x;
    const int t = threadIdx.x;

    float s0 = 0.0f, s1 = 0.0f;
    const float* p = partials + (size_t)b * PARTIALS * 2;
    #pragma unroll
    for (int i = t; i < PARTIALS; i += 256) {   // fixed order per thread
        s0 += p[i * 2 + 0];
        s1 += p[i * 2 + 1];
    }
    sm0[t] = s0; sm1[t] = s1;
    __syncthreads();
    #pragma unroll
    for (int off = 128; off >= 1; off >>= 1) {  // fixed reduction tree
        if (t < off) { sm0[t] += sm0[t + off]; sm1[t] += sm1[t + off]; }
        __syncthreads();
    }
    if (t == 0) {
        float mean = sm1[0] / sm0[0];
        scale[b] = mean / (0.5f * (float)M_);
    }
}

// -------- Stage 3: per-row gather along M with per-batch scale --------------
// Indices are monotone near-identity -> reads stay nearly coalesced and the
// 128 MiB input is L2-resident (192 MB) after stage 1.
__global__ __launch_bounds__(256)
void dn_gather(const float* __restrict__ d, const float* __restrict__ scale,
               float* __restrict__ out) {
    const int row = blockIdx.x;            // (b, l) flattened: B_*L_ rows
    const int b   = row / L_;
    const float s = scale[b];

    const float* in_row  = d   + (size_t)row * M_;
    float*       out_row = out + (size_t)row * M_;

    #pragma unroll
    for (int j = 0; j < 4; ++j) {
        const int m   = threadIdx.x + 256 * j;
        int idx = (int)(s * (float)m);     // trunc-toward-zero == astype(int32)
        idx = idx < 0 ? 0 : (idx > (M_ - 1) ? (M_ - 1) : idx);
        out_row[m] = in_row[idx];
    }
}

extern "C" void kernel_launch(void* const* d_in, const int* in_sizes, int n_in,
                              void* d_out, int out_size, void* d_ws, size_t ws_size,
                              hipStream_t stream) {
    (void)in_sizes; (void)n_in; (void)out_size; (void)ws_size;
    const float* d   = (const float*)d_in[0];
    float*       out = (float*)d_out;

    // Workspace layout (floats): [B_*PARTIALS*2] partials, then [B_] scales.
    float* partials = (float*)d_ws;
    float* scale    = partials + (size_t)B_ * PARTIALS * 2;

    // Stage 1: B_*PARTIALS waves, 8 waves (256 threads) per block.
    const int waves  = B_ * PARTIALS;          // 16384
    dn_colsum_wmma<<<waves / 8, 256, 0, stream>>>(d, partials);

    // Stage 2: one block per batch, deterministic tree reduction.
    dn_finalize<<<B_, 256, 0, stream>>>(partials, scale);

    // Stage 3: one block per (b, l) row, 4 elements per thread.
    dn_gather<<<B_ * L_, 256, 0, stream>>>(d, scale, out);
}